// SinkhornDistance_41248865911227
// MI455X (gfx1250) — compile-verified
//
#include <hip/hip_runtime.h>
#include <math.h>

#define BB 8
#define NN 1024
#define MM 1024
#define DD 64
#define EPS_F 0.1f
#define INV_EPS 10.0f

typedef __attribute__((ext_vector_type(2))) float v2f;
typedef __attribute__((ext_vector_type(8))) float v8f;

// ---------------------------------------------------------------------------
// init: zero U/V, precompute log_nu = log(w + 1e-8)
// ---------------------------------------------------------------------------
__global__ void sinkhorn_init(const float* __restrict__ w,
                              float* __restrict__ U, float* __restrict__ V,
                              float* __restrict__ log_nu) {
  int t = blockIdx.x * blockDim.x + threadIdx.x;
  if (t < BB * MM) {
    U[t] = 0.0f;
    V[t] = 0.0f;
    log_nu[t] = __logf(w[t] + 1e-8f);
  }
}

// ---------------------------------------------------------------------------
// Cost matrix: C = ||x||^2 + ||y||^2 - 2 x.y^T via fp32 WMMA (16x16x4).
// One block computes a 64x64 tile of C for one batch. 8 waves, each wave
// does two 16x16 subtiles, K=64 handled as 16 WMMA steps from LDS.
// ---------------------------------------------------------------------------
__global__ void __launch_bounds__(256)
cost_matrix_wmma(const float* __restrict__ x, const float* __restrict__ y,
                 float* __restrict__ C) {
  __shared__ float xs[64][68];   // pad 68: stride%64==4 -> conflict-free cols
  __shared__ float ys[64][68];
  __shared__ float xn[64];
  __shared__ float yn[64];

  const int b  = blockIdx.z;
  const int i0 = blockIdx.y * 64;
  const int j0 = blockIdx.x * 64;
  const int t  = threadIdx.x;

  const float* xb = x + ((size_t)b * NN + i0) * DD;
  const float* yb = y + ((size_t)b * MM + j0) * DD;

  // stage 64x64 tiles of x and y into LDS (float4, fully coalesced)
  {
    const int r = t >> 4;            // 0..15
    const int c = (t & 15) << 2;     // 0,4,...,60
#pragma unroll
    for (int rr = 0; rr < 64; rr += 16) {
      float4 vx = *(const float4*)(xb + (size_t)(r + rr) * DD + c);
      float4 vy = *(const float4*)(yb + (size_t)(r + rr) * DD + c);
      *(float4*)&xs[r + rr][c] = vx;
      *(float4*)&ys[r + rr][c] = vy;
    }
  }
  __syncthreads();

  // squared norms of each staged row
  if (t < 64) {
    float s = 0.f;
#pragma unroll 8
    for (int c = 0; c < DD; ++c) s += xs[t][c] * xs[t][c];
    xn[t] = s;
  } else if (t < 128) {
    const int r = t - 64;
    float s = 0.f;
#pragma unroll 8
    for (int c = 0; c < DD; ++c) s += ys[r][c] * ys[r][c];
    yn[r] = s;
  }
  __syncthreads();

  const int lane  = t & 31;
  const int wv    = t >> 5;             // wave id 0..7
  const int row16 = lane & 15;
  const int koff  = (lane >> 4) << 1;   // 0 or 2  (A/B K split across half-waves)
  const int mbase = (lane >> 4) << 3;   // 0 or 8  (C/D M split across half-waves)

  float* Cb = C + (size_t)b * NN * MM;

#pragma unroll
  for (int sidx = 0; sidx < 2; ++sidx) {
    const int st = wv + sidx * 8;       // subtile 0..15 in 4x4 grid
    const int sm = st >> 2;
    const int sn = st & 3;

    v8f acc = {0.f, 0.f, 0.f, 0.f, 0.f, 0.f, 0.f, 0.f};
#pragma unroll
    for (int k = 0; k < DD; k += 4) {
      v2f a, bf;
      // A 16x4 (MxK): lane row = lane&15, VGPR0/1 = K = koff / koff+1
      a.x  = xs[sm * 16 + row16][k + koff];
      a.y  = xs[sm * 16 + row16][k + koff + 1];
      // B 4x16 (KxN): B[k][n] = y_tile[n][k]
      bf.x = ys[sn * 16 + row16][k + koff];
      bf.y = ys[sn * 16 + row16][k + koff + 1];
      acc = __builtin_amdgcn_wmma_f32_16x16x4_f32(
                false, a, false, bf, (short)0, acc, false, false);
    }

    const int n = sn * 16 + (lane & 15);
#pragma unroll
    for (int v = 0; v < 8; ++v) {
      const int m = sm * 16 + mbase + v;
      const float cv = xn[m] + yn[n] - 2.0f * acc[v];
      Cb[(size_t)(i0 + m) * MM + (j0 + n)] = cv;
    }
  }
}

// ---------------------------------------------------------------------------
// Row half-iteration: U_i = eps*(log_mu - lse_j(M)) + U_i
// Block = 8 rows x 32 lanes (C rows read coalesced, L2-resident).
// ---------------------------------------------------------------------------
__global__ void __launch_bounds__(256)
sinkhorn_row(const float* __restrict__ C, float* __restrict__ U,
             const float* __restrict__ V, float log_mu) {
  const int b  = blockIdx.y;
  const int tx = threadIdx.x & 31;
  const int ty = threadIdx.x >> 5;
  const int i  = blockIdx.x * 8 + ty;

  const float* Crow = C + ((size_t)b * NN + i) * MM;
  const float* Vb   = V + b * MM;
  const float  ui   = U[b * NN + i];

  float ssum = 0.0f;
  for (int j = tx; j < MM; j += 32)
    ssum += __expf((-Crow[j] + ui + Vb[j]) * INV_EPS);

#pragma unroll
  for (int off = 16; off > 0; off >>= 1)
    ssum += __shfl_down(ssum, off, 32);

  if (tx == 0) {
    const float lse = __logf(ssum + 1e-6f);
    U[b * NN + i] = EPS_F * (log_mu - lse) + ui;
  }
}

// ---------------------------------------------------------------------------
// Col half-iteration: V_j = eps*(log_nu_j - lse_i(M^T)) + V_j
// Block = 32 cols x 8-way parallel over i (coalesced 128B rows), LDS reduce.
// ---------------------------------------------------------------------------
__global__ void __launch_bounds__(256)
sinkhorn_col(const float* __restrict__ C, const float* __restrict__ U,
             float* __restrict__ V, const float* __restrict__ log_nu) {
  __shared__ float red[8][32];
  const int b  = blockIdx.y;
  const int tx = threadIdx.x & 31;
  const int ty = threadIdx.x >> 5;
  const int j  = blockIdx.x * 32 + tx;

  const float* Cb = C + (size_t)b * NN * MM;
  const float* Ub = U + b * NN;
  const float  vj = V[b * MM + j];

  float ssum = 0.0f;
  for (int i = ty; i < NN; i += 8)
    ssum += __expf((-Cb[(size_t)i * MM + j] + Ub[i] + vj) * INV_EPS);

  red[ty][tx] = ssum;
  __syncthreads();
  if (ty == 0) {
    float s = red[0][tx];
#pragma unroll
    for (int r = 1; r < 8; ++r) s += red[r][tx];
    const float lse = __logf(s + 1e-6f);
    V[b * MM + j] = EPS_F * (log_nu[b * MM + j] - lse) + vj;
  }
}

// ---------------------------------------------------------------------------
// Final: pi = exp(M), per-block partial of sum(pi*C) -> deterministic reduce.
// ---------------------------------------------------------------------------
__global__ void __launch_bounds__(256)
sinkhorn_final(const float* __restrict__ C, const float* __restrict__ U,
               const float* __restrict__ V, float* __restrict__ pi,
               float* __restrict__ partial) {
  __shared__ float wsum[8];
  const int b  = blockIdx.y;
  const int tx = threadIdx.x & 31;
  const int ty = threadIdx.x >> 5;
  const int i  = blockIdx.x * 8 + ty;

  const float* Crow = C  + ((size_t)b * NN + i) * MM;
  float*       prow = pi + ((size_t)b * NN + i) * MM;
  const float* Vb   = V + b * MM;
  const float  ui   = U[b * NN + i];

  float acc = 0.0f;
  for (int j = tx; j < MM; j += 32) {
    const float cv = Crow[j];
    const float p  = __expf((-cv + ui + Vb[j]) * INV_EPS);
    prow[j] = p;
    acc += p * cv;
  }
#pragma unroll
  for (int off = 16; off > 0; off >>= 1)
    acc += __shfl_down(acc, off, 32);
  if (tx == 0) wsum[ty] = acc;
  __syncthreads();
  if (threadIdx.x == 0) {
    float s = 0.f;
#pragma unroll
    for (int r = 0; r < 8; ++r) s += wsum[r];
    partial[b * 128 + blockIdx.x] = s;
  }
}

__global__ void __launch_bounds__(128)
reduce_cost(const float* __restrict__ partial, float* __restrict__ cost) {
  __shared__ float sh[128];
  const int b = blockIdx.x;
  sh[threadIdx.x] = partial[b * 128 + threadIdx.x];
  __syncthreads();
  for (int s = 64; s > 0; s >>= 1) {
    if (threadIdx.x < s) sh[threadIdx.x] += sh[threadIdx.x + s];
    __syncthreads();
  }
  if (threadIdx.x == 0) cost[b] = sh[0];
}

// ---------------------------------------------------------------------------
// Launch: init -> WMMA cost matrix -> 20x(row,col) -> final -> reduce.
// All on `stream`; C lives in the output buffer and stays L2-resident (32MB).
// ---------------------------------------------------------------------------
extern "C" void kernel_launch(void* const* d_in, const int* in_sizes, int n_in,
                              void* d_out, int out_size, void* d_ws, size_t ws_size,
                              hipStream_t stream) {
  const float* x = (const float*)d_in[0];   // (8,1024,64)
  const float* y = (const float*)d_in[1];   // (8,1024,64)
  const float* w = (const float*)d_in[2];   // (8,1024)

  float* cost = (float*)d_out;                       // (8,)
  float* pi   = cost + 8;                            // (8,1024,1024)
  float* C    = pi + (size_t)BB * NN * MM;           // (8,1024,1024)

  float* U       = (float*)d_ws;                     // (8,1024)
  float* V       = U + BB * NN;                      // (8,1024)
  float* log_nu  = V + BB * MM;                      // (8,1024)
  float* partial = log_nu + BB * MM;                 // (8,128)

  sinkhorn_init<<<(BB * MM + 255) / 256, 256, 0, stream>>>(w, U, V, log_nu);

  dim3 gC(MM / 64, NN / 64, BB);                     // 16 x 16 x 8 tiles
  cost_matrix_wmma<<<gC, 256, 0, stream>>>(x, y, C);

  const float log_mu = logf(1.0f / (float)NN + 1e-8f);
  for (int it = 0; it < 20; ++it) {
    sinkhorn_row<<<dim3(NN / 8, BB), 256, 0, stream>>>(C, U, V, log_mu);
    sinkhorn_col<<<dim3(MM / 32, BB), 256, 0, stream>>>(C, U, V, log_nu);
  }

  sinkhorn_final<<<dim3(NN / 8, BB), 256, 0, stream>>>(C, U, V, pi, partial);
  reduce_cost<<<BB, 128, 0, stream>>>(partial, cost);
}